// PromptedGAT_30339648979300
// MI455X (gfx1250) — compile-verified
//
#include <hip/hip_runtime.h>

typedef float v2f __attribute__((ext_vector_type(2)));
typedef float v8f __attribute__((ext_vector_type(8)));

#define HFEAT 128
#define NEG_SLOPE 0.2f

// ---------------------------------------------------------------------------
// One-time weight transpose: WT[n*K + k] = W[k*H + n].  Makes the WMMA
// B-fragment (W[k][n], W[k+1][n]) a single contiguous 8-byte load.
// ---------------------------------------------------------------------------
__global__ void gat_transpose_w(const float* __restrict__ W, float* __restrict__ WT,
                                int K) {
  const int i = blockIdx.x * blockDim.x + threadIdx.x;
  if (i >= K * HFEAT) return;
  const int k = i / HFEAT;
  const int n = i - k * HFEAT;
  WT[(size_t)n * K + k] = W[i];
}

// ---------------------------------------------------------------------------
// GEMM: Hout[M x 128] = X[M x K] @ W[K x 128], via V_WMMA_F32_16X16X4_F32.
// W is supplied pre-transposed as WT[128 x K].
// One wave -> 16 rows x 128 cols (8 column tiles). 8 waves per block.
// A layout (32-bit A 16x4): lane l holds A[l&15][k0 + 2*(l>>4) + {0,1}]
// B layout (mirror):        lane l holds B[k0 + 2*(l>>4) + {0,1}][tile*16 + (l&15)]
//                           = WT[tile*16 + (l&15)][k0 + 2*(l>>4) + {0,1}] (one b64)
// D layout: acc vgpr r, lanes 0-15 -> row r, lanes 16-31 -> row r+8; col = lane&15
// ---------------------------------------------------------------------------
__global__ __launch_bounds__(256) void gat_gemm_wmma(const float* __restrict__ X,
                                                     const float* __restrict__ WT,
                                                     float* __restrict__ Hout,
                                                     int M, int K) {
  const int lane    = threadIdx.x & 31;
  const int wave    = threadIdx.x >> 5;
  const int rowbase = (blockIdx.x * 8 + wave) * 16;
  if (rowbase >= M) return;

  const int halfsel = lane >> 4;   // 0: lanes 0-15, 1: lanes 16-31
  const int l16     = lane & 15;

  v8f acc[8] = {};

  const float* Xrow = X + (size_t)(rowbase + l16) * (size_t)K;
  const float* Wcol = WT + (size_t)l16 * (size_t)K;   // + tile*16*K per tile

  for (int k0 = 0; k0 < K; k0 += 4) {
    // A fragment: float2 load, 8-byte aligned (k0 + 2*halfsel is even)
    v2f a = *(const v2f*)(Xrow + k0 + 2 * halfsel);
#pragma unroll
    for (int t = 0; t < 8; ++t) {
      // B fragment: single float2 load from transposed weights
      v2f b = *(const v2f*)(Wcol + (size_t)t * 16 * K + k0 + 2 * halfsel);
      acc[t] = __builtin_amdgcn_wmma_f32_16x16x4_f32(
          /*neg_a=*/false, a, /*neg_b=*/false, b,
          /*c_mod=*/(short)0, acc[t], /*reuse_a=*/false, /*reuse_b=*/false);
    }
  }

#pragma unroll
  for (int t = 0; t < 8; ++t) {
#pragma unroll
    for (int r = 0; r < 8; ++r) {
      const int row = rowbase + r + 8 * halfsel;
      Hout[(size_t)row * HFEAT + t * 16 + l16] = acc[t][r];
    }
  }
}

// ---------------------------------------------------------------------------
// Per-node attention scores: el[i] = h[i] . al, er[i] = h[i] . ar
// One wave32 per node; 4 features per lane; butterfly reduce.
// ---------------------------------------------------------------------------
__global__ __launch_bounds__(256) void gat_node_scores(const float* __restrict__ Hin,
                                                       const float* __restrict__ al,
                                                       const float* __restrict__ ar,
                                                       float* __restrict__ el,
                                                       float* __restrict__ er, int M) {
  const int node = (int)((blockIdx.x * blockDim.x + threadIdx.x) >> 5);
  const int lane = threadIdx.x & 31;
  if (node >= M) return;
  const float* h = Hin + (size_t)node * HFEAT;
  float sl = 0.f, sr = 0.f;
#pragma unroll
  for (int j = 0; j < 4; ++j) {
    const int c = lane + 32 * j;
    const float v = h[c];
    sl += v * al[c];
    sr += v * ar[c];
  }
#pragma unroll
  for (int off = 16; off > 0; off >>= 1) {
    sl += __shfl_xor(sl, off, 32);
    sr += __shfl_xor(sr, off, 32);
  }
  if (lane == 0) { el[node] = sl; er[node] = sr; }
}

// ---------------------------------------------------------------------------
// Init per-layer buffers: m = -inf, denom = 0, agg = 0
// ---------------------------------------------------------------------------
__global__ void gat_init(float* __restrict__ m, float* __restrict__ denom,
                         float* __restrict__ agg, int n_nodes) {
  const int i = blockIdx.x * blockDim.x + threadIdx.x;
  if (i < n_nodes) { m[i] = -__builtin_inff(); denom[i] = 0.f; }
  if (i < n_nodes * HFEAT) agg[i] = 0.f;
}

// ---------------------------------------------------------------------------
// Edge pass 1: e = leaky_relu(el[src] + er[dst]); segment max into m[dst]
// ---------------------------------------------------------------------------
__global__ void gat_edge_max(const int* __restrict__ src, const int* __restrict__ dst,
                             const float* __restrict__ el, const float* __restrict__ er,
                             float* __restrict__ ebuf, float* __restrict__ m, int E) {
  const int e = blockIdx.x * blockDim.x + threadIdx.x;
  if (e >= E) return;
  const int s = src[e], d = dst[e];
  float v = el[s] + er[d];
  v = v > 0.f ? v : NEG_SLOPE * v;
  ebuf[e] = v;
  (void)__hip_atomic_fetch_max(&m[d], v, __ATOMIC_RELAXED, __HIP_MEMORY_SCOPE_AGENT);
}

// ---------------------------------------------------------------------------
// Edge pass 2: ex = exp(e - m[dst]); segment sum into denom[dst]
// ---------------------------------------------------------------------------
__global__ void gat_edge_exp(const int* __restrict__ dst, const float* __restrict__ m,
                             float* __restrict__ ebuf, float* __restrict__ denom, int E) {
  const int e = blockIdx.x * blockDim.x + threadIdx.x;
  if (e >= E) return;
  const int d = dst[e];
  const float ex = expf(ebuf[e] - m[d]);
  ebuf[e] = ex;
  atomicAdd(&denom[d], ex);
}

// ---------------------------------------------------------------------------
// Edge pass 3: agg[dst] += (ex/denom[dst]) * h[src]. One wave32 per edge,
// 4 features per lane, coalesced 128B gathers, fp32 global atomics (L2).
// ---------------------------------------------------------------------------
__global__ __launch_bounds__(256) void gat_edge_agg(const int* __restrict__ src,
                                                    const int* __restrict__ dst,
                                                    const float* __restrict__ ebuf,
                                                    const float* __restrict__ denom,
                                                    const float* __restrict__ Hin,
                                                    float* __restrict__ agg, int E) {
  const int e    = (int)((blockIdx.x * blockDim.x + threadIdx.x) >> 5);
  const int lane = threadIdx.x & 31;
  if (e >= E) return;
  const int s = src[e], d = dst[e];
  const float alpha = ebuf[e] / denom[d];
  const float* hs = Hin + (size_t)s * HFEAT;
  float* ad = agg + (size_t)d * HFEAT;
#pragma unroll
  for (int j = 0; j < 4; ++j) {
    const int c = lane + 32 * j;
    atomicAdd(&ad[c], alpha * hs[c]);
  }
}

// ---------------------------------------------------------------------------
// Finalize: out = relu(agg + b)
// ---------------------------------------------------------------------------
__global__ void gat_finalize(const float* __restrict__ agg, const float* __restrict__ b,
                             float* __restrict__ out, int n_nodes) {
  const int i = blockIdx.x * blockDim.x + threadIdx.x;
  if (i >= n_nodes * HFEAT) return;
  const float v = agg[i] + b[i & (HFEAT - 1)];
  out[i] = v > 0.f ? v : 0.f;
}

// ---------------------------------------------------------------------------
// Launcher. Inputs: feat, src, dst, W1, al1, ar1, b1, W2, al2, ar2, b2
// Output: [h1 (N*128) | h2 (N*128)] fp32.
// ---------------------------------------------------------------------------
extern "C" void kernel_launch(void* const* d_in, const int* in_sizes, int n_in,
                              void* d_out, int out_size, void* d_ws, size_t ws_size,
                              hipStream_t stream) {
  const float* feat = (const float*)d_in[0];
  const int*   src  = (const int*)d_in[1];
  const int*   dst  = (const int*)d_in[2];
  const float* W1   = (const float*)d_in[3];
  const float* al1  = (const float*)d_in[4];
  const float* ar1  = (const float*)d_in[5];
  const float* b1   = (const float*)d_in[6];
  const float* W2   = (const float*)d_in[7];
  const float* al2  = (const float*)d_in[8];
  const float* ar2  = (const float*)d_in[9];
  const float* b2   = (const float*)d_in[10];

  const int H  = in_sizes[4];             // 128
  const int K1 = in_sizes[3] / H;         // 512
  const int N  = in_sizes[0] / K1;        // 100000
  const int E  = in_sizes[1];             // 1600000

  float* h1 = (float*)d_out;                  // N*H
  float* h2 = (float*)d_out + (size_t)N * H;  // N*H

  // Workspace layout (floats): wt[K1*H] el[N] er[N] m[N] denom[N] ebuf[E] agg[N*H]
  float* wt    = (float*)d_ws;
  float* el    = wt + (size_t)K1 * H;
  float* er    = el + N;
  float* mx    = er + N;
  float* denom = mx + N;
  float* ebuf  = denom + N;
  float* agg   = ebuf + E;

  const int rowTiles   = (N + 15) / 16;
  const int gemmBlocks = (rowTiles + 7) / 8;
  const int nodeWaveBl = (N * 32 + 255) / 256;          // wave-per-node kernels
  const int featBlocks = (int)(((size_t)N * H + 255) / 256);
  const int edgeBlocks = (E + 255) / 256;
  const int edgeWaveBl = (int)(((size_t)E * 32 + 255) / 256); // wave-per-edge
  const int tr1Blocks  = (K1 * H + 255) / 256;
  const int tr2Blocks  = (H * H + 255) / 256;

  // ---- Layer 1 ----
  gat_transpose_w<<<tr1Blocks, 256, 0, stream>>>(W1, wt, K1);
  gat_gemm_wmma<<<gemmBlocks, 256, 0, stream>>>(feat, wt, h1, N, K1);
  gat_node_scores<<<nodeWaveBl, 256, 0, stream>>>(h1, al1, ar1, el, er, N);
  gat_init<<<featBlocks, 256, 0, stream>>>(mx, denom, agg, N);
  gat_edge_max<<<edgeBlocks, 256, 0, stream>>>(src, dst, el, er, ebuf, mx, E);
  gat_edge_exp<<<edgeBlocks, 256, 0, stream>>>(dst, mx, ebuf, denom, E);
  gat_edge_agg<<<edgeWaveBl, 256, 0, stream>>>(src, dst, ebuf, denom, h1, agg, E);
  gat_finalize<<<featBlocks, 256, 0, stream>>>(agg, b1, h1, N); // overwrites staging

  // ---- Layer 2 (input = h1, staging in h2 region) ----
  gat_transpose_w<<<tr2Blocks, 256, 0, stream>>>(W2, wt, H);
  gat_gemm_wmma<<<gemmBlocks, 256, 0, stream>>>(h1, wt, h2, N, H);
  gat_node_scores<<<nodeWaveBl, 256, 0, stream>>>(h2, al2, ar2, el, er, N);
  gat_init<<<featBlocks, 256, 0, stream>>>(mx, denom, agg, N);
  gat_edge_max<<<edgeBlocks, 256, 0, stream>>>(src, dst, el, er, ebuf, mx, E);
  gat_edge_exp<<<edgeBlocks, 256, 0, stream>>>(dst, mx, ebuf, denom, E);
  gat_edge_agg<<<edgeWaveBl, 256, 0, stream>>>(src, dst, ebuf, denom, h2, agg, E);
  gat_finalize<<<featBlocks, 256, 0, stream>>>(agg, b2, h2, N);
}